// energyfunc_loss_23845658428155
// MI455X (gfx1250) — compile-verified
//
#include <hip/hip_runtime.h>

typedef __attribute__((ext_vector_type(2))) float v2f;
typedef __attribute__((ext_vector_type(8))) float v8f;
typedef __attribute__((ext_vector_type(4))) unsigned int u32x4;
typedef __attribute__((ext_vector_type(8))) int i32x8;
typedef __attribute__((ext_vector_type(4))) int i32x4;

#define NPTS   1024
#define DCOLS  2050
#define SIGMA  10.0f
// alpha = (1/L^2) * log2(e) = 4 * 1.4426950408889634
#define ALPHA  5.770780163555854f
#define TWOA   (2.0f * ALPHA)

#if __has_include(<hip/amd_detail/amd_gfx1250_TDM.h>)
#define TDM_6ARG 1
#endif

// ws layout (floats):
//   px      [0      , 65536)
//   py      [65536  , 131072)
//   pz      [131072 , 196608)
//   rs      [196608 , 262144)   rs[i] = ALPHA * |p_i|^2
//   partial [262144 , 264192)   one per tile-block (2048)
// px/py/pz/rs are contiguous at stride 65536 floats -> a (256 x 4) TDM tile
// with tensor_dim0_stride = 65536 stages all four 256-column slices at once.

// ---------------------------------------------------------------------------
// Kernel 1: build positions (random-walk cumsum) via LDS parallel scan.
// ---------------------------------------------------------------------------
__global__ __launch_bounds__(1024)
void pos_kernel(const float* __restrict__ conf,
                float* __restrict__ px, float* __restrict__ py,
                float* __restrict__ pz, float* __restrict__ rs) {
    __shared__ float sx[1024], sy[1024], sz[1024];
    const int b = blockIdx.x;
    const int t = threadIdx.x;
    const float* row = conf + b * DCOLS;

    float ex, ey, ez;
    if (t == 0) {
        ex = row[1]; ey = row[2]; ez = row[3];   // initial position seeds the scan
    } else {
        float r1 = row[4 + 2 * (t - 1)] + 0.5f;  // in [0,1]
        float r2 = row[5 + 2 * (t - 1)] + 0.5f;  // in [0,1]
        float cth = 1.0f - 2.0f * r2;            // cos(theta)
        // sin(theta) = sqrt(1-cth^2) = 2*sqrt(r2*(1-r2)); theta in [0,pi]
        float st = 2.0f * __builtin_amdgcn_sqrtf(fmaxf(r2 - r2 * r2, 0.0f));
        // hardware trig takes revolutions: sinf_hw(x) = sin(2*pi*x)
        ex = st * __builtin_amdgcn_cosf(r1);
        ey = st * __builtin_amdgcn_sinf(r1);
        ez = cth;
    }

    sx[t] = ex; sy[t] = ey; sz[t] = ez;
    __syncthreads();

    for (int off = 1; off < 1024; off <<= 1) {   // Hillis-Steele inclusive scan
        float ax = 0.0f, ay = 0.0f, az = 0.0f;
        if (t >= off) { ax = sx[t - off]; ay = sy[t - off]; az = sz[t - off]; }
        __syncthreads();
        sx[t] += ax; sy[t] += ay; sz[t] += az;
        __syncthreads();
    }

    float x = sx[t], y = sy[t], z = sz[t];
    int gi = b * NPTS + t;
    px[gi] = x; py[gi] = y; pz[gi] = z;
    rs[gi] = ALPHA * (x * x + y * y + z * z);
}

// ---------------------------------------------------------------------------
// Kernel 2: block = (batch, iseg, jseg). 8 waves = 8 consecutive i-tiles,
// all sweeping the same 256 j-columns, TDM-staged into LDS once per block.
// Inner loop: 3 ds_load (immediate offsets, no selects) + 1 WMMA + exp2 chain.
// Diagonal tile handled on a readfirstlane-scalarized slow path only.
// ---------------------------------------------------------------------------
__global__ __launch_bounds__(256)
void tile_kernel(const float* __restrict__ px, const float* __restrict__ py,
                 const float* __restrict__ pz, const float* __restrict__ rs,
                 float* __restrict__ partial) {
    // [0:256)=x [256:512)=y [512:768)=z [768:1024)=rs [1024:1280)=zeros
    __shared__ float stage[1280];
    __shared__ float wsum[8];

    const int lane  = threadIdx.x & 31;
    const int wave  = threadIdx.x >> 5;
    const int blk   = blockIdx.x;        // 0..2047
    const int batch = blk >> 5;          // 64
    const int iseg  = (blk >> 2) & 7;    // 8
    const int jseg  = blk & 3;           // 4
    const int base  = batch * NPTS;
    const int j0    = jseg * 256;

    stage[1024 + threadIdx.x] = 0.0f;    // zero pad read by hi lanes as B.y (K=3)

#if defined(__has_builtin) && __has_builtin(__builtin_amdgcn_tensor_load_to_lds)
    // --- TDM staging: one 256x4 f32 tile (rows px,py,pz,rs @ stride 65536) ---
    if (wave == 0) {
        unsigned long long ga = (unsigned long long)(uintptr_t)(px + base + j0);
        unsigned lds_off = (unsigned)(uintptr_t)(&stage[0]);  // LDS aperture: low 32 bits = LDS offset

        u32x4 g0;
        g0.x = 0x1u;                                            // count=1, user descriptor
        g0.y = lds_off;                                         // lds_addr
        g0.z = (unsigned)ga;                                    // global_addr[31:0]
        g0.w = ((unsigned)(ga >> 32) & 0x01FFFFFFu) | (2u << 30); // global_addr[56:32] | type=2

        i32x8 g1 = { 0x00020000,   // workgroup_mask=0, data_size=2 (4B)
                     0x01000000,   // tensor_dim0=256 -> bits[63:48]
                     0x00040000,   // tensor_dim1=4   -> bits[95:80]
                     0x01000000,   // tile_dim0=256   -> bits[127:112]
                     0x00000004,   // tile_dim1=4, tile_dim2=0
                     0x00010000,   // tensor_dim0_stride=65536 -> bits[191:160]
                     0, 0 };       // stride hi / tensor_dim1_stride = 0
        i32x4 g2 = { 0, 0, 0, 0 }; // 2D: dims 2..4 unused
        i32x4 g3 = { 0, 0, 0, 0 };
#ifdef TDM_6ARG
        i32x8 g4 = { 0, 0, 0, 0, 0, 0, 0, 0 };
        __builtin_amdgcn_tensor_load_to_lds(g0, g1, g2, g3, g4, 0);
#else
        __builtin_amdgcn_tensor_load_to_lds(g0, g1, g2, g3, 0);
#endif
        __builtin_amdgcn_s_wait_tensorcnt(0);
    }
#else
    { // fallback: plain staging
        int col = threadIdx.x;
        int g = base + j0 + col;
        stage[col]       = px[g];
        stage[256 + col] = py[g];
        stage[512 + col] = pz[g];
        stage[768 + col] = rs[g];
    }
#endif
    __syncthreads();

    const int  ti  = iseg * 8 + wave;
    const int  bi  = base + ti * 16;
    const int  l15 = lane & 15;
    const bool hiL = lane >= 16;

    // A tile 16x4 f32, pre-scaled by 2*alpha (unconditional loads + selects):
    // lanes 0-15: v0=K0(x), v1=K1(y); lanes 16-31: v0=K2(z), v1=K3(0)
    float axv = px[bi + l15];
    float ayv = py[bi + l15];
    float azv = pz[bi + l15];
    v2f a;
    a.x = TWOA * (hiL ? azv : axv);
    a.y = hiL ? 0.0f : (TWOA * ayv);

    // rs_i for this lane-half's 8 rows (32B-aligned: bi%16==0)
    const int mbase = bi + (hiL ? 8 : 0);
    const float4* rsv = (const float4*)(rs + mbase);
    float4 r0 = rsv[0];
    float4 r1 = rsv[1];
    float rsi[8] = { r0.x, r0.y, r0.z, r0.w, r1.x, r1.y, r1.z, r1.w };

    // Per-lane LDS bases: no selects inside the loop. Hi lanes read z where
    // low lanes read x, and read the zero pad where low lanes read y.
    const float* pA = &stage[(hiL ? 512 : 0) + l15];      // B.v0: x / z
    const float* pB = &stage[(hiL ? 1024 : 256) + l15];   // B.v1: y / 0
    const float* pR = &stage[768 + l15];                  // rs_j

    // Diagonal bookkeeping (forced into an SGPR so the branch is scalar):
    const int tdiag = __builtin_amdgcn_readfirstlane(
        ((ti >> 4) == jseg) ? (ti & 15) : -1);
    const int dv = hiL ? (l15 - 8) : l15;                 // valid only in [0,8)

    float sum = 0.0f;
#pragma unroll 8
    for (int tt = 0; tt < 16; ++tt) {
        v2f bm;
        bm.x = pA[tt * 16];
        bm.y = pB[tt * 16];
        float rsj = pR[tt * 16];

        v8f c = {};
        v8f d = __builtin_amdgcn_wmma_f32_16x16x4_f32(
            /*neg_a=*/false, a, /*neg_b=*/false, bm,
            /*c_mod=*/(short)0, c, /*reuse_a=*/false, /*reuse_b=*/false);

        if (tt == tdiag) {   // scalar (SGPR) compare -> s_cbranch
#pragma unroll
            for (int v = 0; v < 8; ++v) {
                float ex2 = d[v] - rsi[v] - rsj;                  // = -alpha*d2
                float e = (ex2 < 0.0f && v != dv) ? __builtin_amdgcn_exp2f(ex2) : 0.0f;
                sum += e;
            }
        } else {             // fast path: no integer masking
#pragma unroll
            for (int v = 0; v < 8; ++v) {
                float ex2 = d[v] - rsi[v] - rsj;                  // = -alpha*d2
                float e = (ex2 < 0.0f) ? __builtin_amdgcn_exp2f(ex2) : 0.0f;
                sum += e;
            }
        }
    }

    // wave32 reduction, then block reduction -> one deterministic partial
    for (int off = 16; off > 0; off >>= 1) sum += __shfl_xor(sum, off);
    if (lane == 0) wsum[wave] = sum;
    __syncthreads();
    if (threadIdx.x == 0) {
        float s = 0.0f;
#pragma unroll
        for (int w = 0; w < 8; ++w) s += wsum[w];
        partial[blk] = s;
    }
}

// ---------------------------------------------------------------------------
// Kernel 3: deterministic fixed-order reduction of 2048 partials.
// ---------------------------------------------------------------------------
__global__ __launch_bounds__(1024)
void reduce_kernel(const float* __restrict__ partial, float* __restrict__ out) {
    __shared__ float s[1024];
    const int t = threadIdx.x;
    float v = partial[t] + partial[t + 1024];
    s[t] = v;
    __syncthreads();
    for (int off = 512; off > 0; off >>= 1) {
        if (t < off) s[t] += s[t + off];
        __syncthreads();
    }
    if (t == 0) out[0] = SIGMA * s[0];
}

// ---------------------------------------------------------------------------
extern "C" void kernel_launch(void* const* d_in, const int* in_sizes, int n_in,
                              void* d_out, int out_size, void* d_ws, size_t ws_size,
                              hipStream_t stream) {
    const float* conf = (const float*)d_in[0];
    float* ws = (float*)d_ws;
    float* px      = ws;
    float* py      = ws + 65536;
    float* pz      = ws + 131072;
    float* rs      = ws + 196608;
    float* partial = ws + 262144;

    pos_kernel<<<64, 1024, 0, stream>>>(conf, px, py, pz, rs);
    tile_kernel<<<2048, 256, 0, stream>>>(px, py, pz, rs, partial);
    reduce_kernel<<<1, 1024, 0, stream>>>(partial, (float*)d_out);
}